// WeightedL1Loss_35227321762100
// MI455X (gfx1250) — compile-verified
//
#include <hip/hip_runtime.h>
#include <math.h>

// ---------------------------------------------------------------------------
// WeightedL1Loss fused kernel for MI455X (gfx1250, wave32).
// Memory-bound: 132 MB streamed @ 23.3 TB/s -> ~5.7us floor; secondary bound
// is the TRANS pipe (12 transcendentals/pixel), so all exp2/log2 use the
// native v_exp_f32/v_log_f32 builtins and all piecewise selects are
// branchless (v_cndmask, no exec-mask divergence in the hot loop).
//  - one pass, float4 (b128) coalesced loads of pred/target/rgb planes
//  - 64x64 weight table + bin edges staged in LDS (17KB of 320KB/WGP)
//  - per-thread accumulators; wave reduce via V_WMMA_F32_16X16X4_F32
//    (A = ones => D row0 = column sums = v[n]+v[n+16]) + 4 shfl_xor steps
//  - per-block partials to scratch; second 1-block kernel reduces
//    deterministically (no float atomics -> stable across graph replays)
// ---------------------------------------------------------------------------

typedef __attribute__((ext_vector_type(2))) float v2f;
typedef __attribute__((ext_vector_type(8))) float v8f;

#define NBINS        64
#define HW           (384 * 384)      /* 147456 */
#define NBATCH       32
#define NPIX         (NBATCH * HW)    /* 4718592 */
#define NQ           (NPIX / 4)       /* 1179648 quads */
#define THREADS      256
#define MAIN_BLOCKS  1152             /* 1152*256*4 == NQ exactly */

// Native transcendentals: v_log_f32 is log2 on AMDGPU; v_exp_f32 is exp2.
__device__ __forceinline__ float fast_log2(float x) {
  return __builtin_amdgcn_logf(x);
}
__device__ __forceinline__ float fast_exp2(float x) {
  return __builtin_amdgcn_exp2f(x);
}

__device__ __forceinline__ float srgb_to_lin(float c) {
  // where(c > 0.04045, ((c+0.055)/1.055)^2.4, c/12.92)
  // On the pow path the argument is > 0.0905 (normal, positive) -> native
  // v_log_f32/v_exp_f32 are safe and single-instruction.
  const float p = fast_exp2(2.4f * fast_log2(__builtin_fmaf(c, 1.0f / 1.055f,
                                                            0.055f / 1.055f)));
  const float l = c * (1.0f / 12.92f);
  return (c > 0.04045f) ? p : l;
}

__device__ __forceinline__ float lab_f(float t) {
  // where(t > 0.008856, cbrt(t), 7.787*t + 16/116); cbrt path has t > 0.008856.
  const float cb = fast_exp2((1.0f / 3.0f) * fast_log2(t));
  const float ln = __builtin_fmaf(7.787f, t, 16.0f / 116.0f);
  return (t > 0.008856f) ? cb : ln;
}

// searchsorted(edges, x, 'right') - 1, clipped to [0, NBINS-1].
// Uniform linspace -> direct estimate + branchless +/-1 fixup vs LDS edges.
// (x >= e[j+1]) and (x < e[j]) are mutually exclusive on sorted edges.
__device__ __forceinline__ int find_bin(float x, const float* __restrict__ e,
                                        float e0, float inv_step) {
  float fj = floorf((x - e0) * inv_step);
  int j = (int)fminf(fmaxf(fj, 0.0f), (float)(NBINS - 1));
  j += (x >= e[j + 1]) ? 1 : 0;
  j -= (x < e[j]) ? 1 : 0;
  return min(max(j, 0), NBINS - 1);
}

// Full-wave (32 lane) reduction of two accumulators.
// WMMA f32 16x16x4 with A == ones: D[m][n] = sum_k B[k][n]. With the value in
// B VGPR0 only (b.y = 0): B[0][n] = v[lane n], B[2][n] = v[lane n+16] =>
// D row0 (lane L, vgpr0) = v[L%16] + v[L%16 + 16]. Finish 16 columns with
// xor-shuffles. Requires EXEC all-ones (callers guarantee full reconvergence).
__device__ __forceinline__ void wave_reduce2(float& w, float& r) {
  v2f a;  a.x = 1.0f; a.y = 1.0f;
  v2f bw; bw.x = w;   bw.y = 0.0f;
  v2f br; br.x = r;   br.y = 0.0f;
  v8f z = {};
  v8f dw = __builtin_amdgcn_wmma_f32_16x16x4_f32(false, a, false, bw, (short)0,
                                                 z, false, false);
  v8f dr = __builtin_amdgcn_wmma_f32_16x16x4_f32(false, a, false, br, (short)0,
                                                 z, false, false);
  float cw = dw[0];
  float cr = dr[0];
#pragma unroll
  for (int off = 1; off <= 8; off <<= 1) {
    cw += __shfl_xor(cw, off, 32);
    cr += __shfl_xor(cr, off, 32);
  }
  w = cw;
  r = cr;
}

__device__ __forceinline__ float elem(const float4& v, int j) {
  return j == 0 ? v.x : (j == 1 ? v.y : (j == 2 ? v.z : v.w));
}

__global__ void __launch_bounds__(THREADS)
wl1_main_kernel(const float* __restrict__ pred, const float* __restrict__ targ,
                const float* __restrict__ rgb, const float* __restrict__ wts,
                const float* __restrict__ ea, const float* __restrict__ eb,
                float* __restrict__ partials) {
  __shared__ float sW[NBINS * NBINS];   // 16 KB weight table
  __shared__ float sEA[NBINS + 1];
  __shared__ float sEB[NBINS + 1];
  __shared__ float sRed[2][THREADS / 32];

  // Stage weight table (b128 loads) and edges into LDS.
  const float4* w4 = reinterpret_cast<const float4*>(wts);
  float4* sW4 = reinterpret_cast<float4*>(sW);
#pragma unroll
  for (int i = threadIdx.x; i < (NBINS * NBINS) / 4; i += THREADS)
    sW4[i] = w4[i];
  for (int i = threadIdx.x; i < NBINS + 1; i += THREADS) {
    sEA[i] = ea[i];
    sEB[i] = eb[i];
  }
  __syncthreads();

  const float eA0 = sEA[0], invA = 1.0f / (sEA[1] - sEA[0]);
  const float eB0 = sEB[0], invB = 1.0f / (sEB[1] - sEB[0]);

  const int tid = blockIdx.x * THREADS + threadIdx.x;
  const int nthreads = gridDim.x * THREADS;

  float wsum = 0.0f;  // sum of l1_map * pixel_weight
  float rsum = 0.0f;  // sum of |da| + |db|

  for (int q = tid; q < NQ; q += nthreads) {
    const int b = q / (HW / 4);
    const int hw = (q - b * (HW / 4)) * 4;

    const float4 pa0 = *reinterpret_cast<const float4*>(pred + (b * 2 + 0) * HW + hw);
    const float4 pa1 = *reinterpret_cast<const float4*>(pred + (b * 2 + 1) * HW + hw);
    const float4 ta0 = *reinterpret_cast<const float4*>(targ + (b * 2 + 0) * HW + hw);
    const float4 ta1 = *reinterpret_cast<const float4*>(targ + (b * 2 + 1) * HW + hw);
    const float4 rr = *reinterpret_cast<const float4*>(rgb + (b * 3 + 0) * HW + hw);
    const float4 gg = *reinterpret_cast<const float4*>(rgb + (b * 3 + 1) * HW + hw);
    const float4 bb = *reinterpret_cast<const float4*>(rgb + (b * 3 + 2) * HW + hw);

#pragma unroll
    for (int j = 0; j < 4; ++j) {
      const float d0 = fabsf(elem(pa0, j) - elem(ta0, j));
      const float d1 = fabsf(elem(pa1, j) - elem(ta1, j));
      rsum += d0 + d1;
      const float l1 = 0.5f * (d0 + d1);

      const float rl = srgb_to_lin(elem(rr, j));
      const float gl = srgb_to_lin(elem(gg, j));
      const float bl = srgb_to_lin(elem(bb, j));

      const float X = 0.412453f * rl + 0.357580f * gl + 0.180423f * bl;
      const float Y = 0.212671f * rl + 0.715160f * gl + 0.072169f * bl;
      const float Z = 0.019334f * rl + 0.119193f * gl + 0.950227f * bl;

      const float fx = lab_f(X * (1.0f / 0.95047f));
      const float fy = lab_f(Y);
      const float fz = lab_f(Z * (1.0f / 1.08883f));

      const float aC = 500.0f * (fx - fy);
      const float bC = 200.0f * (fy - fz);

      const int ab = find_bin(aC, sEA, eA0, invA);
      const int bbn = find_bin(bC, sEB, eB0, invB);
      wsum += l1 * sW[ab * NBINS + bbn];
    }
  }

  // Wave reduce (WMMA + shfl), then block reduce in LDS.
  wave_reduce2(wsum, rsum);
  const int wave = threadIdx.x >> 5;
  const int lane = threadIdx.x & 31;
  if (lane == 0) {
    sRed[0][wave] = wsum;
    sRed[1][wave] = rsum;
  }
  __syncthreads();
  if (threadIdx.x == 0) {
    float W = 0.0f, R = 0.0f;
#pragma unroll
    for (int i = 0; i < THREADS / 32; ++i) {
      W += sRed[0][i];
      R += sRed[1][i];
    }
    partials[2 * blockIdx.x + 0] = W;
    partials[2 * blockIdx.x + 1] = R;
  }
}

__global__ void __launch_bounds__(THREADS)
wl1_final_kernel(const float* __restrict__ partials, float* __restrict__ out) {
  __shared__ float sRed[2][THREADS / 32];
  float W = 0.0f, R = 0.0f;
  for (int i = threadIdx.x; i < MAIN_BLOCKS; i += THREADS) {
    W += partials[2 * i + 0];
    R += partials[2 * i + 1];
  }
  wave_reduce2(W, R);
  const int wave = threadIdx.x >> 5;
  const int lane = threadIdx.x & 31;
  if (lane == 0) {
    sRed[0][wave] = W;
    sRed[1][wave] = R;
  }
  __syncthreads();
  if (threadIdx.x == 0) {
    float tw = 0.0f, tr = 0.0f;
#pragma unroll
    for (int i = 0; i < THREADS / 32; ++i) {
      tw += sRed[0][i];
      tr += sRed[1][i];
    }
    out[0] = tw / (float)NPIX;           // weighted_l1_loss
    out[1] = tr / (float)(2 * NPIX);     // raw_l1_metric
  }
}

extern "C" void kernel_launch(void* const* d_in, const int* in_sizes, int n_in,
                              void* d_out, int out_size, void* d_ws,
                              size_t ws_size, hipStream_t stream) {
  const float* pred = (const float*)d_in[0];  // [32,2,384,384]
  const float* targ = (const float*)d_in[1];  // [32,2,384,384]
  const float* rgb  = (const float*)d_in[2];  // [32,3,384,384]
  const float* wts  = (const float*)d_in[3];  // [64,64]
  const float* ea   = (const float*)d_in[4];  // [65]
  const float* eb   = (const float*)d_in[5];  // [65]
  float* out = (float*)d_out;                 // [2]
  float* partials = (float*)d_ws;             // 2*MAIN_BLOCKS floats

  hipLaunchKernelGGL(wl1_main_kernel, dim3(MAIN_BLOCKS), dim3(THREADS), 0,
                     stream, pred, targ, rgb, wts, ea, eb, partials);
  hipLaunchKernelGGL(wl1_final_kernel, dim3(1), dim3(THREADS), 0, stream,
                     partials, out);
}